// GATConv_56556129354465
// MI455X (gfx1250) — compile-verified
//
#include <hip/hip_runtime.h>
#include <hip/hip_bf16.h>

// GAT forward for MI455X (gfx1250), wave32.
// Pipeline:
//   1) gat_gemm      : h = feat @ W            (fp32 WMMA 16x16x4)
//   2) gat_logits    : el/er = <h, attn_l/r>   per (node, head)
//   3) gat_init_out  : out[n, c] = bias[c]
//   4) gat_edge_max  : segment max via monotone-encoded u32 atomicMax
//   5) gat_fin_max   : decode, guard non-finite -> 0
//   6) gat_edge_esum : denom[dst,h] += exp(leaky(el[src]+er[dst]) - emax[dst])
//   7) gat_scatter   : out[dst] += a * h[src]  (fp32 hardware atomics, L2-resident)

typedef __attribute__((ext_vector_type(2))) float v2f;
typedef __attribute__((ext_vector_type(8))) float v8f;

#define GAT_FIN   256
#define GAT_HD    256
#define GAT_H     4
#define GAT_D     64
#define NEG_SLOPE 0.2f

// ---------------------------------------------------------------- 1) GEMM ---
// Block = 256 threads = 8 waves. Each wave computes a 16x64 strip of h:
// 4 accumulators (16x16 each) share one A fragment per K-step.
// Grid.x covers ceil(N/32) row-tiles (waves 0..3 -> rows [0,16), 4..7 -> [16,32)).
__global__ __launch_bounds__(256)
void gat_gemm(const float* __restrict__ feat, const float* __restrict__ Wm,
              float* __restrict__ h, int N) {
    const int lane = threadIdx.x & 31;
    const int wave = threadIdx.x >> 5;
    const int row0 = blockIdx.x * 32 + (wave >> 2) * 16;
    const int col0 = (wave & 3) * 64;

    const int m     = lane & 15;          // row within A tile / col within B,C tiles
    const int khalf = (lane >> 4) * 2;    // lanes 0-15: K+{0,1}; lanes 16-31: K+{2,3}

    int row  = row0 + m;
    int rowc = row < N ? row : N - 1;     // clamp loads; EXEC stays all-ones for WMMA

    v8f acc[4];
#pragma unroll
    for (int j = 0; j < 4; ++j) acc[j] = (v8f){0.f,0.f,0.f,0.f,0.f,0.f,0.f,0.f};

    const float* __restrict__ arow = feat + (size_t)rowc * GAT_FIN;

    for (int k0 = 0; k0 < GAT_FIN; k0 += 4) {
        v2f a = *(const v2f*)(arow + k0 + khalf);     // one global_load_b64
#pragma unroll
        for (int j = 0; j < 4; ++j) {
            const float* __restrict__ bp =
                Wm + (size_t)(k0 + khalf) * GAT_HD + (col0 + j * 16 + m);
            v2f b;
            b.x = bp[0];
            b.y = bp[GAT_HD];
            acc[j] = __builtin_amdgcn_wmma_f32_16x16x4_f32(
                false, a, false, b, (short)0, acc[j], false, false);
        }
    }

    // C/D layout: VGPR i holds row (i + 8*(lane>=16)), col = lane&15
    const int rbase = row0 + ((lane >> 4) ? 8 : 0);
    const int ccol  = lane & 15;
    float* __restrict__ cbase = h + (size_t)rbase * GAT_HD + col0 + ccol;

    if (row0 + 16 <= N) {
        // full tile: 32 unguarded coalesced stores with immediate offsets
#pragma unroll
        for (int j = 0; j < 4; ++j)
#pragma unroll
            for (int i = 0; i < 8; ++i)
                cbase[(size_t)i * GAT_HD + j * 16] = acc[j][i];
    } else {
        // ragged tail tile (never taken for N % 32 == 0, kept for generality)
#pragma unroll
        for (int j = 0; j < 4; ++j)
#pragma unroll
            for (int i = 0; i < 8; ++i)
                if (rbase + i < N)
                    cbase[(size_t)i * GAT_HD + j * 16] = acc[j][i];
    }
}

// ------------------------------------------------------------- 2) logits ----
__global__ void gat_logits(const float* __restrict__ h,
                           const float* __restrict__ attn_l,
                           const float* __restrict__ attn_r,
                           float* __restrict__ el, float* __restrict__ er, int N) {
    int idx = blockIdx.x * blockDim.x + threadIdx.x;   // n*H + hh
    if (idx >= N * GAT_H) return;
    int n  = idx >> 2;
    int hh = idx & 3;
    const float4* hp = (const float4*)(h + (size_t)n * GAT_HD + hh * GAT_D);
    const float4* al = (const float4*)(attn_l + hh * GAT_D);
    const float4* ar = (const float4*)(attn_r + hh * GAT_D);
    float sl = 0.f, sr = 0.f;
#pragma unroll 4
    for (int d = 0; d < GAT_D / 4; ++d) {
        float4 v = hp[d], l = al[d], r = ar[d];
        sl += v.x * l.x + v.y * l.y + v.z * l.z + v.w * l.w;
        sr += v.x * r.x + v.y * r.y + v.z * r.z + v.w * r.w;
    }
    el[idx] = sl;
    er[idx] = sr;
}

// ----------------------------------------------------------- 3) init out ----
__global__ void gat_init_out(float* __restrict__ out, const float* __restrict__ bias,
                             long long total) {
    long long i = (long long)blockIdx.x * blockDim.x + threadIdx.x;
    if (i < total) out[i] = bias[i & (GAT_HD - 1)];
}

// ------------------------------------------------------------ helpers -------
__device__ __forceinline__ float leaky(float x) {
    return x > 0.f ? x : NEG_SLOPE * x;
}
// Monotone order-preserving float->u32 key
__device__ __forceinline__ unsigned f2key(float f) {
    unsigned u = __float_as_uint(f);
    return (u & 0x80000000u) ? ~u : (u | 0x80000000u);
}
__device__ __forceinline__ float key2f(unsigned k) {
    unsigned u = (k & 0x80000000u) ? (k & 0x7FFFFFFFu) : ~k;
    return __uint_as_float(u);
}

// ----------------------------------------------------------- 4) seg max -----
__global__ void gat_edge_max(const int* __restrict__ src, const int* __restrict__ dst,
                             const float* __restrict__ el, const float* __restrict__ er,
                             unsigned* __restrict__ emax_enc, int E) {
    int e = blockIdx.x * blockDim.x + threadIdx.x;
    if (e >= E) return;
    int s = src[e], d = dst[e];
    float4 elv = ((const float4*)el)[s];     // one b128 per node row
    float4 erv = ((const float4*)er)[d];
    float x0 = leaky(elv.x + erv.x), x1 = leaky(elv.y + erv.y);
    float x2 = leaky(elv.z + erv.z), x3 = leaky(elv.w + erv.w);
    atomicMax(&emax_enc[d * GAT_H + 0], f2key(x0));
    atomicMax(&emax_enc[d * GAT_H + 1], f2key(x1));
    atomicMax(&emax_enc[d * GAT_H + 2], f2key(x2));
    atomicMax(&emax_enc[d * GAT_H + 3], f2key(x3));
}

// ------------------------------------------------------- 5) finalize max ----
__global__ void gat_fin_max(unsigned* __restrict__ emax_enc, int NH) {
    int i = blockIdx.x * blockDim.x + threadIdx.x;
    if (i >= NH) return;
    unsigned k = emax_enc[i];
    float f = (k == 0u) ? 0.f : key2f(k);      // k==0 -> node never touched
    if (!isfinite(f)) f = 0.f;
    ((float*)emax_enc)[i] = f;                 // in-place decode
}

// ---------------------------------------------------------- 6) exp sums -----
__global__ void gat_edge_esum(const int* __restrict__ src, const int* __restrict__ dst,
                              const float* __restrict__ el, const float* __restrict__ er,
                              const float* __restrict__ emax, float* __restrict__ denom,
                              int E) {
    int e = blockIdx.x * blockDim.x + threadIdx.x;
    if (e >= E) return;
    int s = src[e], d = dst[e];
    float4 elv = ((const float4*)el)[s];
    float4 erv = ((const float4*)er)[d];
    float4 emv = ((const float4*)emax)[d];
    float w0 = __expf(leaky(elv.x + erv.x) - emv.x);
    float w1 = __expf(leaky(elv.y + erv.y) - emv.y);
    float w2 = __expf(leaky(elv.z + erv.z) - emv.z);
    float w3 = __expf(leaky(elv.w + erv.w) - emv.w);
    unsafeAtomicAdd(&denom[d * GAT_H + 0], w0);
    unsafeAtomicAdd(&denom[d * GAT_H + 1], w1);
    unsafeAtomicAdd(&denom[d * GAT_H + 2], w2);
    unsafeAtomicAdd(&denom[d * GAT_H + 3], w3);
}

// ------------------------------------------------------------ 7) scatter ----
// One block (256 threads) per edge; threads 0..3 compute the 4 per-head softmax
// weights into LDS, then all 256 threads stream h[src]*a into out[dst] with
// hardware fp32 atomics (out is 102 MB -> L2-resident on the 192 MB L2).
__global__ __launch_bounds__(256)
void gat_scatter(const int* __restrict__ src, const int* __restrict__ dst,
                 const float* __restrict__ el, const float* __restrict__ er,
                 const float* __restrict__ emax, const float* __restrict__ denom,
                 const float* __restrict__ h, float* __restrict__ out, int E) {
    __shared__ float a_sh[GAT_H];
    int e = blockIdx.x;
    if (e >= E) return;
    int s = src[e], d = dst[e];
    int t = threadIdx.x;
    if (t < GAT_H) {
        float x = leaky(el[s * GAT_H + t] + er[d * GAT_H + t]);
        float w = __expf(x - emax[d * GAT_H + t]);
        a_sh[t] = w / fmaxf(denom[d * GAT_H + t], 1e-16f);
    }
    __syncthreads();
    float a = a_sh[t >> 6];                       // head = t / D
    float v = h[(size_t)s * GAT_HD + t] * a;
    unsafeAtomicAdd(&out[(size_t)d * GAT_HD + t], v);
}

// ---------------------------------------------------------------------------
extern "C" void kernel_launch(void* const* d_in, const int* in_sizes, int n_in,
                              void* d_out, int out_size, void* d_ws, size_t ws_size,
                              hipStream_t stream) {
    const float* feat   = (const float*)d_in[0];
    const float* Wm     = (const float*)d_in[1];
    const float* attn_l = (const float*)d_in[2];
    const float* attn_r = (const float*)d_in[3];
    const float* bias   = (const float*)d_in[4];
    const int*   src    = (const int*)d_in[5];
    const int*   dst    = (const int*)d_in[6];

    const int HD  = in_sizes[4];             // 256
    const int FIN = in_sizes[1] / HD;        // 256
    const int N   = in_sizes[0] / FIN;       // 100000
    const int E   = in_sizes[5];             // 1600000
    const int NH  = N * GAT_H;

    float* out = (float*)d_out;

    // workspace layout
    float*    h     = (float*)d_ws;                          // N * 256
    float*    el    = h  + (size_t)N * GAT_HD;               // N * 4
    float*    er    = el + (size_t)NH;                       // N * 4
    float*    emax  = er + (size_t)NH;                       // N * 4 (u32 keys then f32)
    float*    denom = emax + (size_t)NH;                     // N * 4
    unsigned* emax_enc = (unsigned*)emax;

    hipMemsetAsync(emax_enc, 0, (size_t)NH * sizeof(unsigned), stream);
    hipMemsetAsync(denom,    0, (size_t)NH * sizeof(float),    stream);

    // 1) GEMM: each block covers 32 rows x 256 cols
    gat_gemm<<<dim3((N + 31) / 32), dim3(256), 0, stream>>>(feat, Wm, h, N);

    // 2) per-node attention logits
    gat_logits<<<(NH + 255) / 256, 256, 0, stream>>>(h, attn_l, attn_r, el, er, N);

    // 3) out = bias (also clears the poisoned output)
    long long total = (long long)N * GAT_HD;
    gat_init_out<<<(unsigned)((total + 255) / 256), 256, 0, stream>>>(out, bias, total);

    // 4) segment max over dst
    gat_edge_max<<<(E + 255) / 256, 256, 0, stream>>>(src, dst, el, er, emax_enc, E);

    // 5) decode + guard
    gat_fin_max<<<(NH + 255) / 256, 256, 0, stream>>>(emax_enc, NH);

    // 6) softmax denominators
    gat_edge_esum<<<(E + 255) / 256, 256, 0, stream>>>(src, dst, el, er, emax, denom, E);

    // 7) weighted scatter-add of source messages
    gat_scatter<<<E, 256, 0, stream>>>(src, dst, el, er, emax, denom, h, out, E);
}